// GiddLoss_41489384079560
// MI455X (gfx1250) — compile-verified
//
#include <hip/hip_runtime.h>
#include <float.h>
#include <math.h>

// ---------------------------------------------------------------------------
// GIDD loss for MI455X (gfx1250).
//
// Data-movement plan: each 128 KB logits row is DMA'd ONCE into LDS by the
// Tensor Data Mover (tensor_load_to_lds, TENSORcnt-tracked), then three
// reduction passes (max, sum-exp, sum-log-p) run out of LDS. HBM traffic is
// 256 MB total -> ~11 us at 23.3 TB/s, which is the roofline for this op.
// Full-row LDS staging is only possible because CDNA5 has 320 KB LDS per WGP;
// two resident workgroups per WGP let one block's DMA overlap the other's
// compute. No matmul structure exists here, so WMMA is inapplicable; the
// CDNA5 feature that pays is the TDM async path + split wait counters.
// ---------------------------------------------------------------------------

typedef float        f32x4 __attribute__((ext_vector_type(4)));
typedef unsigned int u32x4 __attribute__((ext_vector_type(4)));
typedef unsigned int u32x8 __attribute__((ext_vector_type(8)));

#define VOCAB        32000
#define MASK_IDX     31999
#define TPB          320            // 10 wave32 waves
#define NWAVES       (TPB / 32)
#define V4           (VOCAB / 4)    // 8000 float4 per row
#define CHUNKS       (V4 / TPB)     // 25 float4 per thread
#define LOG_CLIP_MIN (-1000000.0f)

__device__ __forceinline__ float waveMax(float v) {
#pragma unroll
  for (int o = 16; o > 0; o >>= 1) v = fmaxf(v, __shfl_xor(v, o, 32));
  return v;
}
__device__ __forceinline__ float waveSum(float v) {
#pragma unroll
  for (int o = 16; o > 0; o >>= 1) v += __shfl_xor(v, o, 32);
  return v;
}

// Block reduction across 10 waves; every thread returns the full result.
__device__ __forceinline__ float blockReduce(float v, bool is_max, float* sred, int tid) {
  v = is_max ? waveMax(v) : waveSum(v);
  if ((tid & 31) == 0) sred[tid >> 5] = v;
  __syncthreads();
  float r = sred[0];
  if (is_max) {
#pragma unroll
    for (int w = 1; w < NWAVES; ++w) r = fmaxf(r, sred[w]);
  } else {
#pragma unroll
    for (int w = 1; w < NWAVES; ++w) r += sred[w];
  }
  __syncthreads();   // sred may be reused by the next reduction
  return r;
}

__global__ __launch_bounds__(TPB)
void gidd_token_kernel(const float* __restrict__ logits,
                       const int*   __restrict__ input_ids,
                       const int*   __restrict__ z_t,
                       const float* __restrict__ t_arr,
                       const float* __restrict__ attn_mask,
                       float* __restrict__ elbo_out,
                       float* __restrict__ loss_terms,
                       int T)
{
  // 16-byte aligned dynamic LDS (32000 floats = 128 KB) so ds_load_b128 and
  // the TDM destination are naturally aligned.
  extern __shared__ f32x4 lds4[];
  float* lds = (float*)lds4;
  __shared__ float sred[NWAVES];

  const int s   = blockIdx.x;             // token index in [0, B*T)
  const int tid = threadIdx.x;
  const int b   = s / T;

  // ---- TDM: DMA the entire logits row into LDS with one instruction ------
  if (tid == 0) {
    unsigned long long ga =
        (unsigned long long)(const void*)(logits + (unsigned long long)s * VOCAB);
    // Low 32 bits of the flat LDS pointer == LDS byte offset (aperture layout).
    unsigned lds_base = (unsigned)(unsigned long long)(const void*)lds4;

    u32x4 g0; u32x8 g1; u32x4 g2; u32x4 g3;
    // D# group 0: count=1 (valid user descriptor), lds_addr, global_addr, type=2
    g0[0] = 1u;
    g0[1] = lds_base;
    g0[2] = (unsigned)(ga & 0xFFFFFFFFull);
    g0[3] = (unsigned)((ga >> 32) & 0x01FFFFFFull) | (2u << 30);
    // D# group 1: wg_mask=0 (not in cluster), data_size=4B, 2D tile 32000 x 1
    g1[0] = (2u << 16);                            // data_size = 4 bytes
    g1[1] = ((unsigned)VOCAB & 0xFFFFu) << 16;     // tensor_dim0[15:0]
    g1[2] = ((unsigned)VOCAB >> 16) | (1u << 16);  // tensor_dim0[31:16] | tensor_dim1=1
    g1[3] = ((unsigned)VOCAB) << 16;               // tensor_dim1[31:16]=0 | tile_dim0=32000
    g1[4] = 1u;                                    // tile_dim1=1, tile_dim2=0 (unused)
    g1[5] = (unsigned)VOCAB;                       // tensor_dim0_stride[31:0]
    g1[6] = ((unsigned)VOCAB & 0xFFFFu) << 16;     // stride0[47:32]=0 | stride1[15:0]
    g1[7] = 0u;                                    // stride1[47:16]
    // Benign higher dims (length 1, tile dims 0 = unused)
    g2[0] = 1u; g2[1] = 1u; g2[2] = 0u; g2[3] = 0u;
    g3[0] = 0u; g3[1] = (1u << 16); g3[2] = 0u; g3[3] = 0u;

    asm volatile("tensor_load_to_lds %0, %1, %2, %3"
                 :: "s"(g0), "s"(g1), "s"(g2), "s"(g3)
                 : "memory");
  }
#if __has_builtin(__builtin_amdgcn_s_wait_tensorcnt)
  __builtin_amdgcn_s_wait_tensorcnt(0);
#else
  asm volatile("s_wait_tensorcnt 0x0" ::: "memory");
#endif
  __syncthreads();

  // ---- per-token schedule scalars (uniform) -------------------------------
  const float tb    = t_arr[b];
  const float Bc    = expf(-11.0f);       // B_ = exp(LOG_B)
  const float gamma = 1.0f;               // exp(LOG_GAMMA)
  const float t1m   = 1.0f - tb;
  const float t_g   = powf(tb,  gamma);
  const float t1m_g = powf(t1m, gamma);
  const float c_t   = sqrtf(t_g) * sqrtf(t1m_g) * Bc;
  const float C_t   = t_g + t1m_g + (float)(VOCAB - 2) * c_t;
  const float alpha_hat = t1m_g - c_t;

  const f32x4* row = lds4;

  // ---- pass 1: row max (MASK logit is -FLT_MAX in the reference) ----------
  float lmax = -FLT_MAX;
#pragma unroll 5
  for (int c = 0; c < CHUNKS; ++c) {
    int i4 = c * TPB + tid;
    f32x4 v = row[i4];
    if (i4 == V4 - 1) v[3] = -FLT_MAX;    // element 31999 == MASK
    lmax = fmaxf(lmax, fmaxf(fmaxf(v[0], v[1]), fmaxf(v[2], v[3])));
  }
  const float m = blockReduce(lmax, true, sred, tid);

  // ---- pass 2: softmax denominator ---------------------------------------
  float lsum = 0.0f;
#pragma unroll 5
  for (int c = 0; c < CHUNKS; ++c) {
    int i4 = c * TPB + tid;
    f32x4 v = row[i4];
    float e3 = (i4 == V4 - 1) ? 0.0f : __expf(v[3] - m);
    lsum += __expf(v[0] - m) + __expf(v[1] - m) + __expf(v[2] - m) + e3;
  }
  const float S    = blockReduce(lsum, false, sred, tid);
  const float invS = 1.0f / S;

  // ---- pass 3: SB = sum_{v != MASK} log(alpha_hat * xhat_v + c_t) ---------
  const float ai = alpha_hat * invS;
  float lsb = 0.0f;
#pragma unroll 5
  for (int c = 0; c < CHUNKS; ++c) {
    int i4 = c * TPB + tid;
    f32x4 v = row[i4];
    float a0 = __logf(fmaf(ai, __expf(v[0] - m), c_t));
    float a1 = __logf(fmaf(ai, __expf(v[1] - m), c_t));
    float a2 = __logf(fmaf(ai, __expf(v[2] - m), c_t));
    float a3 = (i4 == V4 - 1) ? 0.0f : __logf(fmaf(ai, __expf(v[3] - m), c_t));
    lsb += (a0 + a1) + (a2 + a3);
  }
  const float SB = blockReduce(lsb, false, sred, tid);

  // ---- finalize token (scalar, precise libm) ------------------------------
  if (tid == 0) {
    const int id = input_ids[s];
    const int z  = z_t[s];

    const float t_g_p   = gamma * powf(tb, gamma - 1.0f);
    const float t1m_g_p = -powf(t1m, gamma - 1.0f);
    const float c_t_p   = 0.5f * gamma * (1.0f - 2.0f * tb) / (tb * t1m) * c_t;
    const float C_t_p   = t_g_p + t1m_g_p + (float)(VOCAB - 2) * c_t_p;
    const float alpha_hat_p = t1m_g_p - c_t_p;

    const float logC = logf(C_t);

    // KL(q_t || p_t) using constant structure of q_t; MASK term is exactly 0
    // because q(MASK) == p(MASK) == t_g/C (xhat(MASK)=0, x(MASK)=0).
    const float l_id  = lds[id];                          // id != MASK always
    const float xh_id = expf(l_id - m) * invS;
    const float lp_id = fmaxf(logf(fmaf(alpha_hat, xh_id, c_t)) - logC, LOG_CLIP_MIN);
    const float qc    = c_t / C_t;
    const float lqc   = fmaxf(logf(qc), LOG_CLIP_MIN);
    const float q_id  = (alpha_hat + c_t) / C_t;
    const float lq_id = fmaxf(logf(q_id), LOG_CLIP_MIN);
    const float sum_lp = SB - (float)(VOCAB - 1) * logC;  // sum_{v!=MASK} log p(v)
    const float kl = (float)(VOCAB - 2) * qc * lqc
                   - qc * (sum_lp - lp_id)
                   + q_id * (lq_id - lp_id);

    // correction term at z_t
    float lq_z, lp_z;
    if (z == MASK_IDX) {
      const float lm_ = fmaxf(logf(t_g / C_t), LOG_CLIP_MIN);
      lq_z = lm_; lp_z = lm_;
    } else {
      const float l_z  = lds[z];
      const float xh_z = expf(l_z - m) * invS;
      lp_z = fmaxf(logf(fmaf(alpha_hat, xh_z, c_t)) - logC, LOG_CLIP_MIN);
      const float qz = ((z == id) ? alpha_hat : 0.0f) + c_t;
      lq_z = fmaxf(logf(qz) - logC, LOG_CLIP_MIN);
    }
    const float log_ratio  = lq_z - lp_z;
    const float correction = -log_ratio + expf(log_ratio);

    // weights (_get_weights)
    const float is_mask  = (z == MASK_IDX) ? 1.0f : 0.0f;
    const float pi_hat   = t_g   * is_mask + c_t   * (1.0f - is_mask);
    const float pi_hat_p = t_g_p * is_mask + c_t_p * (1.0f - is_mask);
    const float alpha    = alpha_hat / C_t;
    const float pi_beta  = pi_hat   / C_t;
    const float alpha_ratio = alpha_hat_p / alpha_hat - C_t_p / C_t;
    const float omega_t  = (pi_hat_p - alpha_hat_p / alpha_hat * pi_hat) / C_t;
    const float is_x     = (z == id) ? 1.0f : 0.0f;
    const float elbo_w   = (1.0f - is_x) * (omega_t / pi_beta)
                         + is_x * (omega_t / (alpha + pi_beta));
    const float loss_w   = fminf(fmaxf(elbo_w, 0.0f), 100.0f);

    const float tot = kl + correction;
    elbo_out[s]   = elbo_w * tot + alpha_ratio;
    loss_terms[s] = loss_w * tot * attn_mask[s];
  }
}

// Deterministic final reduction: loss = sum(loss_terms) / sum(mask).
__global__ __launch_bounds__(256)
void gidd_finalize_kernel(const float* __restrict__ terms,
                          const float* __restrict__ mask,
                          int n, float* __restrict__ out0)
{
  __shared__ float sa[8], sb[8];
  float a = 0.0f, bm = 0.0f;
  for (int i = threadIdx.x; i < n; i += 256) { a += terms[i]; bm += mask[i]; }
  a  = waveSum(a);
  bm = waveSum(bm);
  if ((threadIdx.x & 31) == 0) { sa[threadIdx.x >> 5] = a; sb[threadIdx.x >> 5] = bm; }
  __syncthreads();
  if (threadIdx.x == 0) {
    float ta = 0.0f, tb = 0.0f;
#pragma unroll
    for (int w = 0; w < 8; ++w) { ta += sa[w]; tb += sb[w]; }
    out0[0] = ta / tb;
  }
}

extern "C" void kernel_launch(void* const* d_in, const int* in_sizes, int n_in,
                              void* d_out, int out_size, void* d_ws, size_t ws_size,
                              hipStream_t stream) {
  const float* logits    = (const float*)d_in[0];
  const int*   input_ids = (const int*)  d_in[1];
  const float* attn_mask = (const float*)d_in[2];
  const int*   z_t       = (const int*)  d_in[3];
  const float* t_arr     = (const float*)d_in[4];

  const int BT = in_sizes[1];       // B*T tokens
  const int B  = in_sizes[4];
  const int T  = BT / B;

  float* out   = (float*)d_out;     // out[0] = loss, out[1..BT] = elbo
  float* terms = (float*)d_ws;      // BT per-token loss terms

  const size_t shmem = (size_t)VOCAB * sizeof(float);   // 128 KB row in LDS
  gidd_token_kernel<<<dim3(BT), dim3(TPB), shmem, stream>>>(
      logits, input_ids, z_t, t_arr, attn_mask, out + 1, terms, T);
  gidd_finalize_kernel<<<dim3(1), dim3(256), 0, stream>>>(terms, attn_mask, BT, out);
}